// MeanSquaredError2GAN_4887672783315
// MI455X (gfx1250) — compile-verified
//
#include <hip/hip_runtime.h>
#include <math.h>

#define COLN 14

typedef __attribute__((ext_vector_type(2))) float v2f;
typedef __attribute__((ext_vector_type(4))) float v4f;
typedef __attribute__((ext_vector_type(8))) float v8f;

// ---------------------------------------------------------------------------
// Kernel 1 (1 block, 256 threads): build the 196 possible normalized target
// maps T[yi*14+xi][196].
//  - thread 0 builds the Gaussian-reflect matrix K in double (matches numpy
//    float64 -> float32) into LDS
//  - each wave handles (yi,xi) pairs: f = outer(K[:,yi],K[:,xi]) via one
//    V_WMMA_F32_16X16X4_F32 (A = col0-only, B = row0-only -> exact fp32
//    rank-1 product, bitwise equal to the reference einsum on a one-hot),
//    true min/max via wave shfl reduction, exact IEEE divide per element.
// ---------------------------------------------------------------------------
__global__ __launch_bounds__(256)
void mse2gan_build_table(float* __restrict__ T) {
    __shared__ float sK[196];
    if (threadIdx.x == 0) {
        double w[9];
        double s = 0.0;
        for (int d = -4; d <= 4; ++d) { w[d + 4] = exp(-0.5 * (double)d * (double)d); s += w[d + 4]; }
        for (int i = 0; i < 9; ++i) w[i] /= s;
        double K[COLN][COLN];
        for (int i = 0; i < COLN; ++i)
            for (int j = 0; j < COLN; ++j) K[i][j] = 0.0;
        for (int i = 0; i < COLN; ++i)
            for (int d = -4; d <= 4; ++d) {
                int j = i + d;
                if (j < 0) j = -j - 1;                 // scipy 'reflect'
                else if (j >= COLN) j = 2 * COLN - 1 - j;
                K[i][j] += w[d + 4];
            }
        for (int i = 0; i < COLN; ++i)
            for (int j = 0; j < COLN; ++j) sK[i * COLN + j] = (float)K[i][j];
    }
    __syncthreads();

    const int wid  = threadIdx.x >> 5;
    const int lane = threadIdx.x & 31;
    const int p    = lane & 15;          // A/B row index (lanes 0-15)
    const int N    = lane & 15;          // C/D column
    const int hi   = lane >> 4;          // C/D row-half
    const bool lo  = (lane < 16) && (p < COLN);

    for (int pair = wid; pair < 196; pair += 8) {      // wave-uniform loop
        const int yi = pair / COLN;
        const int xi = pair % COLN;
        v2f A;  A[0]  = lo ? sK[p * COLN + yi] : 0.0f;  A[1]  = 0.0f;
        v2f Bm; Bm[0] = lo ? sK[p * COLN + xi] : 0.0f;  Bm[1] = 0.0f;
        v8f c = {};
        c = __builtin_amdgcn_wmma_f32_16x16x4_f32(
                false, A, false, Bm, (short)0, c, false, false);

        // exact min/max over the 14x14 region
        float lmn = 3.4e38f, lmx = -3.4e38f;
#pragma unroll
        for (int r = 0; r < 8; ++r) {
            const int M = r + hi * 8;
            if (M < COLN && N < COLN) { const float f = c[r]; lmn = fminf(lmn, f); lmx = fmaxf(lmx, f); }
        }
        for (int off = 16; off > 0; off >>= 1) {
            lmn = fminf(lmn, __shfl_xor(lmn, off, 32));
            lmx = fmaxf(lmx, __shfl_xor(lmx, off, 32));
        }
        const float denom = fmaxf(lmx - lmn, 1e-12f);

        float* Trow = T + (size_t)pair * 196;
#pragma unroll
        for (int r = 0; r < 8; ++r) {
            const int M = r + hi * 8;
            if (M < COLN && N < COLN)
                Trow[M * COLN + N] = (c[r] - lmn) / denom;   // exact IEEE divide (38K total)
        }
    }
}

// ---------------------------------------------------------------------------
// Kernel 2: streaming kernel, one wave32 per (b,j) tile. Table rows and h are
// 16B-aligned -> b128 loads; tt stores are dword (out+1 is 4B-aligned). The
// vis branch is wave-uniform -> no per-element predication, no divides.
// ---------------------------------------------------------------------------
__global__ __launch_bounds__(256)
void mse2gan_main(const float* __restrict__ h, const float* __restrict__ t,
                  const int* __restrict__ v, const float* __restrict__ T,
                  float* __restrict__ out, float* __restrict__ partials, int bjTotal) {
    __shared__ float sRed[8];
    const int tid  = threadIdx.x;
    const int wid  = tid >> 5;
    const int lane = tid & 31;
    const int tile = blockIdx.x * 8 + wid;

    float acc = 0.0f;
    if (tile < bjTotal) {                              // wave-uniform guard
        const float tx = t[tile * 2 + 0];
        const float ty = t[tile * 2 + 1];
        const int xi = (int)truncf(tx * 14.0f);
        const int yi = (int)truncf(ty * 14.0f);
        const bool valid = (xi >= 0) & (xi <= 13) & (yi >= 0) & (yi <= 13);
        const bool vis = (v[tile * 2 + 0] == 1) & valid;

        const v4f* h4 = (const v4f*)(h + (size_t)tile * 196);
        float* oT = out + 1 + (size_t)tile * 196;

        if (vis) {                                     // wave-uniform branch
            const v4f* T4 = (const v4f*)(T + (size_t)(yi * COLN + xi) * 196);
            for (int q = lane; q < 49; q += 32) {
                const v4f tt4 = T4[q];
                const v4f hh  = h4[q];
                float* o4 = oT + q * 4;
#pragma unroll
                for (int c0 = 0; c0 < 4; ++c0) {
                    o4[c0] = tt4[c0];
                    const float dif = hh[c0] - tt4[c0];
                    acc += dif * dif;
                }
            }
        } else {
            for (int q = lane; q < 49; q += 32) {
                const v4f hh = h4[q];
                float* o4 = oT + q * 4;
#pragma unroll
                for (int c0 = 0; c0 < 4; ++c0) {
                    o4[c0] = 0.0f;
                    const float dif = (q * 4 + c0 < COLN) ? 0.0f : hh[c0]; // row0 zeroed
                    acc += dif * dif;
                }
            }
        }
    }

    for (int off = 16; off > 0; off >>= 1) acc += __shfl_down(acc, off, 32);
    if (lane == 0) sRed[wid] = acc;
    __syncthreads();
    if (tid == 0) {
        float s = 0.0f;
        for (int i = 0; i < 8; ++i) s += sRed[i];      // fixed order
        partials[blockIdx.x] = s;
    }
}

// ---------------------------------------------------------------------------
// Kernel 3: single block. Deterministic double-precision sum of partials +
// integer visible-count reduction; d1 = sum / cnt.
// ---------------------------------------------------------------------------
__global__ __launch_bounds__(1024)
void mse2gan_reduce(const float* __restrict__ partials, int nPart,
                    const float* __restrict__ t, const int* __restrict__ v,
                    int bjTotal, float* __restrict__ out) {
    __shared__ double sD[1024];
    __shared__ int    sI[1024];
    const int tid = threadIdx.x;
    double dacc = 0.0;
    for (int i = tid; i < nPart; i += 1024) dacc += (double)partials[i];
    int cacc = 0;
    for (int i = tid; i < bjTotal; i += 1024) {
        const float tx = t[i * 2 + 0];
        const float ty = t[i * 2 + 1];
        const int xi = (int)truncf(tx * 14.0f);
        const int yi = (int)truncf(ty * 14.0f);
        const bool valid = (xi >= 0) && (xi <= 13) && (yi >= 0) && (yi <= 13);
        if (valid && v[i * 2 + 0] == 1) cacc++;
    }
    sD[tid] = dacc;
    sI[tid] = cacc;
    __syncthreads();
    for (int s = 512; s > 0; s >>= 1) {
        if (tid < s) { sD[tid] += sD[tid + s]; sI[tid] += sI[tid + s]; }
        __syncthreads();
    }
    if (tid == 0) out[0] = (float)sD[0] / (float)sI[0];
}

// ---------------------------------------------------------------------------
extern "C" void kernel_launch(void* const* d_in, const int* in_sizes, int n_in,
                              void* d_out, int out_size, void* d_ws, size_t ws_size,
                              hipStream_t stream) {
    (void)d_in[0]; (void)n_in; (void)out_size; (void)ws_size;  // 'o' unused by reference
    const float* h = (const float*)d_in[1];
    const float* t = (const float*)d_in[2];
    const int*   v = (const int*)d_in[3];
    float* out = (float*)d_out;

    const int bjTotal = in_sizes[1] / (COLN * COLN);           // B * NJ
    float* ws       = (float*)d_ws;
    float* T        = ws;                // 196*196 floats = 150 KB
    float* partials = ws + 38656;        // one float per block

    const int wavesPerBlock = 8;
    const int nBlocks = (bjTotal + wavesPerBlock - 1) / wavesPerBlock;

    mse2gan_build_table<<<1, 256, 0, stream>>>(T);
    mse2gan_main<<<nBlocks, 256, 0, stream>>>(h, t, v, T, out, partials, bjTotal);
    mse2gan_reduce<<<1, 1024, 0, stream>>>(partials, nBlocks, t, v, bjTotal, out);
}